// SpatiallyJitterColorChannels_38302518346462
// MI455X (gfx1250) — compile-verified
//
#include <hip/hip_runtime.h>
#include <stdint.h>

// Problem constants from the reference (fixed shapes).
#define SHIFT 2
#define B_ 64
#define C_ 3
#define H_ 512
#define W_ 512

// gfx1250 async global->LDS copy builtin (tracked by ASYNCcnt).
// Guarded so host-pass / older toolchains fall back to plain b128 staging.
#if defined(__has_builtin)
#if __has_builtin(__builtin_amdgcn_global_load_async_to_lds_b128)
#define USE_ASYNC_LDS 1
#endif
#endif

typedef float f32x4 __attribute__((ext_vector_type(4)));
typedef int   v4i   __attribute__((ext_vector_type(4)));

__global__ __launch_bounds__(128) void SpatiallyJitterColorChannels_38302518346462_kernel(
    const float* __restrict__ x, const int* __restrict__ shifts, float* __restrict__ out) {
  __shared__ float row[W_];

  const int t  = threadIdx.x;        // 0..127, one float4 of the row each
  const int r  = blockIdx.x;         // flat output row id: (b*C + c)*H + h
  const int h  = r & (H_ - 1);
  const int bc = r >> 9;             // b*C + c  (H = 512 = 2^9)

  // Per-(b,c) shifts, uniform across the block -> scalar loads.
  const int sh = shifts[bc * 2 + 0] - SHIFT;
  const int sw = shifts[bc * 2 + 1] - SHIFT;

  // H-roll: just select the source row.
  const int hsrc = (h - sh) & (H_ - 1);
  const float* __restrict__ src_row = x + ((size_t)bc * H_ + (size_t)hsrc) * W_;
  float* __restrict__ dst_row = out + (size_t)r * W_;

  // ---- Stage the source row into LDS with aligned 128-bit transfers ----
#if USE_ASYNC_LDS
  {
    typedef __attribute__((address_space(1))) v4i g_v4i;
    typedef __attribute__((address_space(3))) v4i l_v4i;
    // Global pointer: inttoptr into AS(1) (flat/global share representation).
    g_v4i* gsrc = (g_v4i*)(uintptr_t)(src_row + 4 * t);
    // LDS pointer: real addrspacecast from the __shared__ generic pointer.
    l_v4i* ldst = (l_v4i*)(&row[4 * t]);
    // (global src, lds dst, imm offset, imm cpol)
    __builtin_amdgcn_global_load_async_to_lds_b128(gsrc, ldst, 0, 0);
    // Drain this wave's ASYNCcnt before the workgroup barrier.
    asm volatile("s_wait_asynccnt 0" ::: "memory");
  }
#else
  *(f32x4*)(&row[4 * t]) = *(const f32x4*)(src_row + 4 * t);
#endif
  __syncthreads();

  // ---- W-roll out of LDS, aligned 128-bit store to global ----
  const int base = (4 * t - sw + W_) & (W_ - 1);
  f32x4 o;
  o.x = row[base];
  o.y = row[(base + 1) & (W_ - 1)];
  o.z = row[(base + 2) & (W_ - 1)];
  o.w = row[(base + 3) & (W_ - 1)];
  *(f32x4*)(dst_row + 4 * t) = o;
}

extern "C" void kernel_launch(void* const* d_in, const int* in_sizes, int n_in,
                              void* d_out, int out_size, void* d_ws, size_t ws_size,
                              hipStream_t stream) {
  const float* x      = (const float*)d_in[0];  // [64,3,512,512] fp32
  const int*   shifts = (const int*)d_in[1];    // [64,3,2] int32
  float*       out    = (float*)d_out;          // [64,3,512,512] fp32

  dim3 grid(B_ * C_ * H_);  // one block per output row (98304 blocks)
  dim3 block(128);          // 4 wave32s; each thread moves one float4
  SpatiallyJitterColorChannels_38302518346462_kernel<<<grid, block, 0, stream>>>(x, shifts, out);
}